// L2_MSE_CLOZE_7421703487839
// MI455X (gfx1250) — compile-verified
//
#include <hip/hip_runtime.h>

// Problem constants from the reference.
#define SDIM 131072
#define DDIM 512
#define VDIM 100000

typedef __attribute__((ext_vector_type(2))) float v2f;
typedef __attribute__((ext_vector_type(8))) float v8f;
typedef __attribute__((ext_vector_type(4))) int   v4i;

#define AS1 __attribute__((address_space(1)))
#define AS3 __attribute__((address_space(3)))

#if __has_builtin(__builtin_amdgcn_wmma_f32_16x16x4_f32)
#define HAVE_WMMA4 1
#endif
#if __has_builtin(__builtin_amdgcn_global_load_async_to_lds_b128)
#define HAVE_ASYNC 1
#endif

__device__ __forceinline__ void wait_asynccnt0() {
#if __has_builtin(__builtin_amdgcn_s_wait_asynccnt)
  __builtin_amdgcn_s_wait_asynccnt(0);
#else
  asm volatile("s_wait_asynccnt 0" ::: "memory");
#endif
}

// ---------------------------------------------------------------------------
// K1: zero the per-vocab counters (ws is poisoned, must re-init every call)
// ---------------------------------------------------------------------------
__global__ void k_zero_counts(int* __restrict__ counts) {
  int i = blockIdx.x * 256 + threadIdx.x;
  if (i < VDIM) counts[i] = 0;
}

// ---------------------------------------------------------------------------
// K2: histogram of masked tokens: counts[l2_data[s]] += (l2_idxs[s]==1)
// ---------------------------------------------------------------------------
__global__ void k_count(const int* __restrict__ data, const int* __restrict__ idxs,
                        int* __restrict__ counts) {
  int s = blockIdx.x * 256 + threadIdx.x;
  if (s < SDIM && idxs[s] == 1) atomicAdd(&counts[data[s]], 1);
}

// ---------------------------------------------------------------------------
// K3: recip[i] = count>0 ? 0.5/count : 0   (per-row scatter scale)
// ---------------------------------------------------------------------------
__global__ void k_recip(const int* __restrict__ counts, float* __restrict__ recip) {
  int i = blockIdx.x * 256 + threadIdx.x;
  if (i < VDIM) {
    int c = counts[i];
    recip[i] = (c > 0) ? (0.5f / (float)c) : 0.0f;
  }
}

// ---------------------------------------------------------------------------
// K4: EMA base init via WMMA.
// One wave per 16x16 tile of [V,D]. Computes D = (0.5*I) x Wtile via four
// chained V_WMMA_F32_16X16X4_F32 (exact in f32: each output element gets a
// single 0.5*x product; remaining FMA terms are 0*y). Then writes
//   out = count>0 ? 0.5*w : w   (w recovered exactly as 2*(0.5*w)).
//
// Layouts (wave32):
//   A 16x4 f32 : lane l, half h=l>=16, m=l-16h; v0 holds K=2h, v1 holds K=2h+1.
//   B 4x16 f32 : by symmetry, v0 holds row K=2h (col = l%16), v1 row K=2h+1.
//   C/D 16x16  : VGPR r, lanes 0-15 -> M=r, lanes 16-31 -> M=r+8, N=l%16.
// ---------------------------------------------------------------------------
__global__ __launch_bounds__(256) void k_ema_init(const float* __restrict__ w,
                                                  const int* __restrict__ counts,
                                                  float* __restrict__ outw) {
  const unsigned lane = threadIdx.x & 31u;
  const unsigned wave = threadIdx.x >> 5;
  const unsigned gw   = blockIdx.x * 8u + wave;     // 0 .. 200000-1 tiles
  const unsigned r0   = (gw >> 5) * 16u;            // vocab-row tile base (6250 tiles)
  const unsigned c0   = (gw & 31u) * 16u;           // D-col tile base (32 tiles)
  const unsigned h    = lane >> 4;                  // half-wave: 0 or 1
  const unsigned n    = lane & 15u;                 // column within tile
  const unsigned m    = lane - 16u * h;             // A-matrix row id for this lane

  v8f acc = {};  // C = 0
#if defined(HAVE_WMMA4)
#pragma unroll
  for (int j = 0; j < 4; ++j) {
    // A_j = rows of (0.5 * Identity), K-slice [4j, 4j+4)
    v2f a;
    a.x = (m == (unsigned)(4 * j + 2 * (int)h))     ? 0.5f : 0.0f;
    a.y = (m == (unsigned)(4 * j + 2 * (int)h + 1)) ? 0.5f : 0.0f;
    // B_j = rows [r0+4j, r0+4j+4) of the W tile (coalesced 64B segments)
    v2f b;
    b.x = w[(size_t)(r0 + 4u * j + 2u * h) * DDIM + c0 + n];
    b.y = w[(size_t)(r0 + 4u * j + 2u * h + 1u) * DDIM + c0 + n];
    acc = __builtin_amdgcn_wmma_f32_16x16x4_f32(false, a, false, b, (short)0, acc,
                                                false, false);
  }
#else
  // Fallback: plain VALU scale (same math, C/D layout).
#pragma unroll
  for (int r = 0; r < 8; ++r)
    acc[r] = 0.5f * w[(size_t)(r0 + (unsigned)r + 8u * h) * DDIM + c0 + n];
#endif

  // Select per vocab row and store (D layout: reg r -> rows r0+r+8h).
#pragma unroll
  for (int r = 0; r < 8; ++r) {
    const unsigned row = r0 + (unsigned)r + 8u * h;
    const float half_w = acc[r];
    const float val = (counts[row] > 0) ? half_w : (half_w + half_w);
    outw[(size_t)row * DDIM + c0 + n] = val;
  }
}

// ---------------------------------------------------------------------------
// K5: scatter. One wave per token row. Masked rows stage their 2KB `out` row
// into LDS via GLOBAL_LOAD_ASYNC_TO_LDS_B128 (4 ops/wave), wait ASYNCcnt,
// then atomically add out[s]*(0.5/count) onto the EMA base in d_out.
// ---------------------------------------------------------------------------
__global__ __launch_bounds__(256) void k_scatter(const float* __restrict__ srcS,
                                                 const int* __restrict__ data,
                                                 const int* __restrict__ idxs,
                                                 const float* __restrict__ recip,
                                                 float* __restrict__ outw) {
  __shared__ float stage[8 * DDIM];  // 16 KB: one 512-float row per wave
  const unsigned lane = threadIdx.x & 31u;
  const unsigned wave = threadIdx.x >> 5;
  const unsigned s    = blockIdx.x * 8u + wave;

  if (idxs[s] != 1) return;  // wave-uniform branch

  const int   id = data[s];
  const float sc = recip[id];  // 0.5 / count (count >= 1 here)
  const float* __restrict__ src = srcS + (size_t)s * DDIM;
  float* __restrict__ lds = stage + wave * DDIM;
  float* __restrict__ dst = outw + (size_t)id * DDIM;

#if defined(HAVE_ASYNC)
#pragma unroll
  for (int q = 0; q < 4; ++q) {
    const unsigned off = (q * 32u + lane) * 4u;  // float index, 16B per lane
    __builtin_amdgcn_global_load_async_to_lds_b128(
        (AS1 v4i*)(src + off), (AS3 v4i*)(lds + off), 0, 0);
  }
  wait_asynccnt0();
#pragma unroll
  for (int q = 0; q < 4; ++q) {
    const unsigned off = (q * 32u + lane) * 4u;
    const float4 v = *(const float4*)(lds + off);
    atomicAdd(&dst[off + 0], v.x * sc);
    atomicAdd(&dst[off + 1], v.y * sc);
    atomicAdd(&dst[off + 2], v.z * sc);
    atomicAdd(&dst[off + 3], v.w * sc);
  }
#else
#pragma unroll
  for (int q = 0; q < 4; ++q) {
    const unsigned off = (q * 32u + lane) * 4u;
    const float4 v = *(const float4*)(src + off);
    atomicAdd(&dst[off + 0], v.x * sc);
    atomicAdd(&dst[off + 1], v.y * sc);
    atomicAdd(&dst[off + 2], v.z * sc);
    atomicAdd(&dst[off + 3], v.w * sc);
  }
#endif
}

// ---------------------------------------------------------------------------
// Launch: counts -> recip -> EMA base (full d_out overwrite) -> atomic scatter.
// Stream ordering provides all inter-kernel dependencies.
// ---------------------------------------------------------------------------
extern "C" void kernel_launch(void* const* d_in, const int* in_sizes, int n_in,
                              void* d_out, int out_size, void* d_ws, size_t ws_size,
                              hipStream_t stream) {
  const float* srcS = (const float*)d_in[0];  // out [S, D]
  const float* w    = (const float*)d_in[1];  // l2_weight [V, D]
  const int*   data = (const int*)d_in[2];    // l2_data [S]
  const int*   idxs = (const int*)d_in[3];    // l2_idxs [S]
  float* outw = (float*)d_out;                // new_weight [V, D]

  int*   counts = (int*)d_ws;                                   // V ints
  float* recip  = (float*)((char*)d_ws + VDIM * sizeof(int));   // V floats

  (void)in_sizes; (void)n_in; (void)out_size; (void)ws_size;

  const int vblocks = (VDIM + 255) / 256;
  k_zero_counts<<<vblocks, 256, 0, stream>>>(counts);
  k_count<<<SDIM / 256, 256, 0, stream>>>(data, idxs, counts);
  k_recip<<<vblocks, 256, 0, stream>>>(counts, recip);
  // 6250 row-tiles * 32 col-tiles = 200000 waves, 8 waves/block
  k_ema_init<<<(VDIM / 16) * (DDIM / 16) / 8, 256, 0, stream>>>(w, counts, outw);
  k_scatter<<<SDIM / 8, 256, 0, stream>>>(srcS, data, idxs, recip, outw);
}